// linear_sig_10746008175152
// MI455X (gfx1250) — compile-verified
//
#include <hip/hip_runtime.h>

#define B_TOTAL 2048
#define T_LEN   128
#define CH      8
#define SIG_DIM 584   // 8 + 64 + 512
#define NOUT    10

typedef float v2f __attribute__((ext_vector_type(2)));
typedef float v8f __attribute__((ext_vector_type(8)));

// ---------------------------------------------------------------------------
// Kernel 1: depth-3 Chen scan. One wave32 per batch element.
// Lane l owns S2 flat [2l, 2l+2) and S3 flat [16l, 16l+16), i.e.
//   a = l>>2, b in {(l&3)*2, (l&3)*2+1}, c = 0..7.
// S1 is never materialized during the scan: S1[t-1] = X[t-1] - X[0].
// ---------------------------------------------------------------------------
__global__ void __launch_bounds__(256)
sig_scan_kernel(const float* __restrict__ X, float* __restrict__ sig) {
  __shared__ float xs[8][T_LEN * CH];   // 8 waves * 4KB = 32KB LDS
  const int lane  = threadIdx.x & 31;
  const int wave  = threadIdx.x >> 5;
  const int batch = blockIdx.x * 8 + wave;

  const float* xb = X + (size_t)batch * (T_LEN * CH);
  float* xw = xs[wave];

  // stage path into LDS: 1024 floats, 8 x float4 per lane
#pragma unroll
  for (int i = 0; i < 8; ++i) {
    const int idx = (i * 32 + lane) * 4;
    *(float4*)(xw + idx) = *(const float4*)(xb + idx);
  }
  __syncthreads();

  const int a  = lane >> 2;
  const int b0 = (lane & 3) << 1;

  float prev[CH], x0[CH];
#pragma unroll
  for (int c = 0; c < CH; ++c) { prev[c] = xw[c]; x0[c] = prev[c]; }
  float preva        = xw[a];
  const float x0a    = preva;
  float prevb0       = xw[b0];
  float prevb1       = xw[b0 + 1];

  float s2j0 = 0.f, s2j1 = 0.f;
  float s3[16];
#pragma unroll
  for (int i = 0; i < 16; ++i) s3[i] = 0.f;

  for (int t = 1; t < T_LEN; ++t) {
    const float* row = xw + t * CH;
    const float4 c0 = *(const float4*)(row);
    const float4 c1 = *(const float4*)(row + 4);
    float cur[CH];
    cur[0]=c0.x; cur[1]=c0.y; cur[2]=c0.z; cur[3]=c0.w;
    cur[4]=c1.x; cur[5]=c1.y; cur[6]=c1.z; cur[7]=c1.w;
    const float cura  = row[a];      // per-lane broadcast-free LDS reads
    const float curb0 = row[b0];
    const float curb1 = row[b0 + 1];

    float dx[CH];
#pragma unroll
    for (int c = 0; c < CH; ++c) dx[c] = cur[c] - prev[c];
    const float dxa = cura  - preva;
    const float db0 = curb0 - prevb0;
    const float db1 = curb1 - prevb1;
    const float s1a = preva - x0a;                // S1_old[a]
    // S3 += S2_old[a,b]*dx[c] + (0.5*S1_old[a] + dx[a]/6) * dx[b]*dx[c]
    const float g   = 0.5f * s1a + (1.f / 6.f) * dxa;
    // S2 += dx[b]*(0.5*dx[a] + S1_old[a])
    const float h   = 0.5f * dxa + s1a;

#pragma unroll
    for (int c = 0; c < CH; ++c) {
      const float t0 = db0 * dx[c];
      s3[c]     = fmaf(s2j0, dx[c], s3[c]);
      s3[c]     = fmaf(g, t0, s3[c]);
      const float t1 = db1 * dx[c];
      s3[8 + c] = fmaf(s2j1, dx[c], s3[8 + c]);
      s3[8 + c] = fmaf(g, t1, s3[8 + c]);
    }
    s2j0 = fmaf(db0, h, s2j0);
    s2j1 = fmaf(db1, h, s2j1);

#pragma unroll
    for (int c = 0; c < CH; ++c) prev[c] = cur[c];
    preva = cura; prevb0 = curb0; prevb1 = curb1;
  }

  float* sb = sig + (size_t)batch * SIG_DIM;
  if (lane == 0) {                                   // level 1: X[T-1]-X[0]
    *(float4*)(sb)     = make_float4(prev[0]-x0[0], prev[1]-x0[1],
                                     prev[2]-x0[2], prev[3]-x0[3]);
    *(float4*)(sb + 4) = make_float4(prev[4]-x0[4], prev[5]-x0[5],
                                     prev[6]-x0[6], prev[7]-x0[7]);
  }
  *(v2f*)(sb + 8 + 2 * lane) = (v2f){s2j0, s2j1};    // level 2
#pragma unroll
  for (int q = 0; q < 4; ++q) {                      // level 3
    *(float4*)(sb + 72 + 16 * lane + 4 * q) =
        make_float4(s3[4*q], s3[4*q+1], s3[4*q+2], s3[4*q+3]);
  }
}

// ---------------------------------------------------------------------------
// Kernel 2: out(2048x10) = sig(2048x584) @ W^T(584x10) + b
// fp32 WMMA 16x16x4, one wave per 16-row tile. EXEC all-1s (blockDim=32).
// A layout: lanes 0-15 (M=lane) hold K={k0,k0+1}; lanes 16-31 hold K={k0+2,k0+3}.
// B layout mirrors with N=lane&15 (rows of W, padded cols masked to zero).
// ---------------------------------------------------------------------------
__global__ void __launch_bounds__(32)
sig_gemm_kernel(const float* __restrict__ sig, const float* __restrict__ W,
                const float* __restrict__ bias, float* __restrict__ out) {
  const int lane = threadIdx.x & 31;
  const int half = lane >> 4;        // K sub-pair select
  const int l16  = lane & 15;        // M for A-frag, N for B-frag
  const int row0 = blockIdx.x * 16;

  const bool  valid_n = (l16 < NOUT);
  const int   n_eff   = valid_n ? l16 : 0;
  const float nmask   = valid_n ? 1.f : 0.f;

  const float* arow = sig + (size_t)(row0 + l16) * SIG_DIM + 2 * half;
  const float* wrow = W   + (size_t)n_eff        * SIG_DIM + 2 * half;

  v8f acc = {};
  for (int k0 = 0; k0 < SIG_DIM; k0 += 4) {
    v2f afrag = *(const v2f*)(arow + k0);
    v2f bfrag = *(const v2f*)(wrow + k0);
    bfrag.x *= nmask;
    bfrag.y *= nmask;
    acc = __builtin_amdgcn_wmma_f32_16x16x4_f32(
        /*neg_a=*/false, afrag, /*neg_b=*/false, bfrag,
        /*c_mod=*/(short)0, acc, /*reuse_a=*/false, /*reuse_b=*/false);
  }

  if (valid_n) {
    const float bv = bias[l16];
#pragma unroll
    for (int r = 0; r < 8; ++r) {
      const int row = row0 + r + half * 8;    // C/D layout: VGPR r -> M=r (+8 hi half)
      out[(size_t)row * NOUT + l16] = acc[r] + bv;
    }
  }
}

extern "C" void kernel_launch(void* const* d_in, const int* in_sizes, int n_in,
                              void* d_out, int out_size, void* d_ws, size_t ws_size,
                              hipStream_t stream) {
  const float* X    = (const float*)d_in[0];   // (2048,128,8) f32
  const float* W    = (const float*)d_in[1];   // (10,584) f32
  const float* bias = (const float*)d_in[2];   // (10,) f32
  float* out = (float*)d_out;                  // (2048,10) f32
  float* sig = (float*)d_ws;                   // 2048*584 f32 scratch (~4.8 MB)

  sig_scan_kernel<<<B_TOTAL / 8, 256, 0, stream>>>(X, sig);
  sig_gemm_kernel<<<B_TOTAL / 16, 32, 0, stream>>>(sig, W, bias, out);
}